// GeneralizedKernelScore_84576495993375
// MI455X (gfx1250) — compile-verified
//
#include <hip/hip_runtime.h>
#include <hip/hip_bf16.h>

typedef __attribute__((ext_vector_type(2))) float v2f;
typedef __attribute__((ext_vector_type(8))) float v8f;

#define DIM 12288
#define MM 8
#define NUM_T 1000
#define WAVES 16
#define KCHUNK (DIM / WAVES)   // 768

__global__ __launch_bounds__(512)
void gks_gram_kernel(const float* __restrict__ x,
                     const float* __restrict__ y,
                     const int*   __restrict__ t,
                     float*       __restrict__ out)
{
    const int n    = blockIdx.x;      // 0..31
    const int tid  = threadIdx.x;     // 0..511
    const int wave = tid >> 5;        // 0..15
    const int lane = tid & 31;

    // WMMA f32 16x16x4 A-layout: lane row = lane%16 holds M (=row of Z);
    // lane half selects K pair: lanes 0-15 -> K {0,1}, lanes 16-31 -> K {2,3}.
    const int row  = lane & 15;                 // Z row: 0..7 = x rows, 8..15 = y rows
    const int kpar = (lane >> 4) << 1;          // 0 or 2

    const float* base = (row < MM)
        ? (x + (size_t)(n * MM + row) * DIM)
        : (y + (size_t)(n * MM + (row - MM)) * DIM);
    const float* p = base + wave * KCHUNK + kpar;

    // Accumulate partial Gram G_partial = Z_chunk * Z_chunk^T in fp32 WMMA.
    v8f acc = {};
    #pragma unroll 8
    for (int k = 0; k < KCHUNK; k += 4) {
        v2f a = *(const v2f*)(p + k);   // Z[row][k0+kpar], Z[row][k0+kpar+1]
        // Gram: A and B operands are the same Z tile (B = A^T in matrix terms,
        // identical per-lane register contents under matching K striping).
        acc = __builtin_amdgcn_wmma_f32_16x16x4_f32(
                  /*neg_a=*/false, a, /*neg_b=*/false, a,
                  /*c_mod=*/(short)0, acc, /*reuse_a=*/false, /*reuse_b=*/false);
    }

    // ---- cross-wave reduction of the 16x16 accumulators ----
    __shared__ float partial[WAVES][256];
    __shared__ float G[256];   // G[m*16 + nn]

    // C/D layout: VGPR r, lanes 0-15 -> M=r, N=lane; lanes 16-31 -> M=r+8, N=lane-16.
    #pragma unroll
    for (int r = 0; r < 8; ++r) {
        const int m = (lane < 16) ? r : (r + 8);
        partial[wave][m * 16 + row] = acc[r];
    }
    __syncthreads();

    if (tid < 256) {
        float s = 0.f;
        #pragma unroll
        for (int w = 0; w < WAVES; ++w) s += partial[w][tid];
        G[tid] = s;
    }
    __syncthreads();

    // ---- final scalars (trivial: 64 exps per n) ----
    if (tid == 0) {
        const float tf   = (float)t[n];
        const float s    = 1.0f - tf / (float)(NUM_T - 1);
        const float beta = 2.0f * s + 0.1f * (1.0f - s);
        const float invD = 1.0f / (float)DIM;

        // confinement: mean_j exp(-beta * ||x_j - y_j||^2 / D)
        float conf = 0.f;
        #pragma unroll
        for (int j = 0; j < MM; ++j) {
            const float d2 = (G[j*16 + j] + G[(MM+j)*16 + (MM+j)]
                              - 2.0f * G[j*16 + (MM+j)]) * invD;
            conf += expf(-beta * d2);
        }
        conf *= (1.0f / (float)MM);

        // interaction: mean over j != jp of exp(-beta * ||x_j - x_jp||^2 / D)
        float inter = 0.f;
        #pragma unroll
        for (int j = 0; j < MM; ++j) {
            #pragma unroll
            for (int jp = 0; jp < MM; ++jp) {
                if (j != jp) {
                    const float d2 = (G[j*16 + j] + G[jp*16 + jp]
                                      - 2.0f * G[j*16 + jp]) * invD;
                    inter += expf(-beta * d2);
                }
            }
        }
        inter *= (1.0f / (float)(MM * (MM - 1)));

        const float im = 0.5f * inter;       // LAMBDA_VAL/2 * interaction
        out[      n] = im - inter * 0.0f - conf;  // score = im - conf
        out[ 32 + n] = conf;
        out[ 64 + n] = inter;
        out[ 96 + n] = im;
    }
}

extern "C" void kernel_launch(void* const* d_in, const int* in_sizes, int n_in,
                              void* d_out, int out_size, void* d_ws, size_t ws_size,
                              hipStream_t stream) {
    const float* x = (const float*)d_in[0];   // (32, 8, 12288) f32
    const float* y = (const float*)d_in[1];   // (32, 8, 12288) f32
    const int*   t = (const int*)d_in[2];     // (32,) i32
    float* out = (float*)d_out;               // 4 x 32 f32, tuple order

    (void)in_sizes; (void)n_in; (void)out_size; (void)d_ws; (void)ws_size;

    gks_gram_kernel<<<32, 512, 0, stream>>>(x, y, t, out);
}